// SurfExtract_58162447123320
// MI455X (gfx1250) — compile-verified
//
#include <hip/hip_runtime.h>
#include <hip/hip_bf16.h>
#include <stdint.h>

// ---------------- problem constants ----------------
#define NPTS   50000
#define KM1    16
#define ROWS   (NPTS * KM1)        // 800000 (point, neighbor) rows
#define MTRI   (ROWS * 2)          // 1600000 triangles
#define SENTK  0x7FFFFFFFFFFFFFFFULL

// sigmoid(x) > t  <=>  x > log(t/(1-t)); compare in sortable-uint domain:
// sortable(+v) = bits(v) ^ 0x80000000 (both thresholds are >= 0)
#define UTH_CONF1 (0x3FB17218u ^ 0x80000000u)  // bits(1.38629436f = ln4) flipped
#define UTH_CONF2 (0x00000000u ^ 0x80000000u)  // bits(0.0f) flipped
#define COS_ANGLE  (-0.5f)                     // cos(120 deg)

// ---------------- radix sort config ----------------
#define RS_BLOCKS  250
#define RS_THREADS 64                // 2 waves (wave32)
#define RS_ITEMS   100               // items per thread (serial => stable)
#define RS_CHUNK   (RS_THREADS * RS_ITEMS)   // 6400; 250*6400 == 1600000
#define RADIX      256
#define PASSES     6                 // bits 0..47 cover all valid keys + SENT low bits

// =====================================================================
// CDNA5 async global->LDS copy (ASYNCcnt path, CDNA5 ISA)
// =====================================================================
__device__ __forceinline__ void async_copy_row64(const float* gsrc, uint32_t lds_off) {
  // INST_OFFSET is added to BOTH global and LDS addresses per the ISA,
  // so one (vdst, vaddr) pair covers all four 16B issues.
  asm volatile("global_load_async_to_lds_b128 %0, %1, off"           :: "v"(lds_off), "v"(gsrc) : "memory");
  asm volatile("global_load_async_to_lds_b128 %0, %1, off offset:16" :: "v"(lds_off), "v"(gsrc) : "memory");
  asm volatile("global_load_async_to_lds_b128 %0, %1, off offset:32" :: "v"(lds_off), "v"(gsrc) : "memory");
  asm volatile("global_load_async_to_lds_b128 %0, %1, off offset:48" :: "v"(lds_off), "v"(gsrc) : "memory");
}
__device__ __forceinline__ void wait_async_all() {
  asm volatile("s_wait_asynccnt 0" ::: "memory");
}

__device__ __forceinline__ uint32_t sortable_f32(float v) {
  uint32_t b = __float_as_uint(v);
  // negative -> ~b ; non-negative -> b | 0x80000000 : strictly monotone map
  uint32_t mask = (uint32_t)(((int32_t)b) >> 31) | 0x80000000u;
  return b ^ mask;
}

__device__ __forceinline__ void sort3i(int& a, int& b, int& c) {
  int t;
  if (a > b) { t = a; a = b; b = t; }
  if (b > c) { t = b; b = c; c = t; }
  if (a > b) { t = a; a = b; b = t; }
}

// =====================================================================
// Stage 1: per (point, neighbor) row -> 2 keys + payloads + triangles
// =====================================================================
__global__ __launch_bounds__(256) void k_stage1(
    const float* __restrict__ points,
    const float* __restrict__ logits,     // (N, 16, 16) row-major
    const int*   __restrict__ knn,        // (N, 17)
    unsigned long long* __restrict__ keys,
    uint32_t* __restrict__ vals,
    int* __restrict__ tris)               // (MTRI, 3) sorted vertex ids
{
  __shared__ float slog[256 * KM1];       // 16 KB: one 16-float row per thread
  const int tid = threadIdx.x;
  const int row = blockIdx.x * 256 + tid; // grid sized exactly: ROWS/256 blocks

  // Stage this thread's 16 logits (64B contiguous) through LDS via ASYNCcnt.
  const float* g = logits + (size_t)blockIdx.x * (256 * KM1) + (size_t)tid * KM1;
  uint32_t loff = (uint32_t)(uintptr_t)(&slog[tid * KM1]); // low 32 bits = LDS offset
  async_copy_row64(g, loff);
  wait_async_all();
  __syncthreads();

  // Branchless stable top-2: key = (sortable_value << 8) | (255 - j).
  // Ties stably prefer lower j (matches argsort(-x) with stable sort).
  unsigned long long best0 = 0ull, best1 = 0ull;
  const float4* rowp = (const float4*)(&slog[tid * KM1]);
#pragma unroll
  for (int q = 0; q < 4; ++q) {
    const float4 r = rowp[q];            // ds_load_b128
    const float vv[4] = { r.x, r.y, r.z, r.w };
#pragma unroll
    for (int c4 = 0; c4 < 4; ++c4) {
      const int j = q * 4 + c4;
      const unsigned long long k =
          ((unsigned long long)sortable_f32(vv[c4]) << 8) | (unsigned long long)(255 - j);
      const unsigned long long hi = (k > best0) ? k : best0;   // v_cmp + cndmask
      const unsigned long long lo = (k > best0) ? best0 : k;
      best0 = hi;
      best1 = (lo > best1) ? lo : best1;
    }
  }
  const int j0 = 255 - (int)(best0 & 0xFF);
  const int j1 = 255 - (int)(best1 & 0xFF);
  const uint32_t u0 = (uint32_t)(best0 >> 8);
  const uint32_t u1 = (uint32_t)(best1 >> 8);
  const bool conf = (u0 > UTH_CONF1) && (u1 > UTH_CONF2);

  const int i  = row >> 4;
  const int e  = row & 15;
  const int kb = i * 17;
  const int c  = knn[kb];
  const int ne = knn[kb + 1 + e];
  const int aA = knn[kb + 1 + j0];
  const int aB = knn[kb + 1 + j1];

  const float pcx = points[3 * c],  pcy = points[3 * c + 1],  pcz = points[3 * c + 2];
  const float pex = points[3 * ne], pey = points[3 * ne + 1], pez = points[3 * ne + 2];
  const float pax = points[3 * aA], pay = points[3 * aA + 1], paz = points[3 * aA + 2];
  const float pbx = points[3 * aB], pby = points[3 * aB + 1], pbz = points[3 * aB + 2];

  const float ex = pex - pcx, ey = pey - pcy, ez = pez - pcz;
  const float ax = pax - pex, ay = pay - pey, az = paz - pez;
  const float bx = pbx - pex, by = pby - pey, bz = pbz - pez;

  // normals = cross(edge, edge_{A,B})
  const float cAx = ey * az - ez * ay, cAy = ez * ax - ex * az, cAz = ex * ay - ey * ax;
  const float cBx = ey * bz - ez * by, cBy = ez * bx - ex * bz, cBz = ex * by - ey * bx;
  const float dAB = cAx * cBx + cAy * cBy + cAz * cBz;
  const float nA  = cAx * cAx + cAy * cAy + cAz * cAz;
  const float nB  = cBx * cBx + cBy * cBy + cBz * cBz;
  const float cosang = dAB * rsqrtf(nA * nB);   // NaN for degenerate -> flip=false, same as ref

  const bool flip  = cosang > COS_ANGLE;
  const bool valid = conf && !flip;

  const size_t base = (size_t)row * 2;
  {
    int x = c, y = ne, z = aA; sort3i(x, y, z);
    unsigned long long key = ((unsigned long long)((long long)x * NPTS + y)) * NPTS + z;
    keys[base] = valid ? key : SENTK;
    vals[base] = (uint32_t)base;
    tris[base * 3 + 0] = x; tris[base * 3 + 1] = y; tris[base * 3 + 2] = z;
  }
  {
    int x = c, y = ne, z = aB; sort3i(x, y, z);
    unsigned long long key = ((unsigned long long)((long long)x * NPTS + y)) * NPTS + z;
    keys[base + 1] = valid ? key : SENTK;
    vals[base + 1] = (uint32_t)(base + 1);
    tris[(base + 1) * 3 + 0] = x; tris[(base + 1) * 3 + 1] = y; tris[(base + 1) * 3 + 2] = z;
  }
}

// =====================================================================
// Radix sort pass kernels (stable LSD, 8-bit digits)
// =====================================================================
__global__ __launch_bounds__(RS_THREADS) void k_hist(
    const unsigned long long* __restrict__ keys, uint32_t* __restrict__ ghist, int shift)
{
  __shared__ unsigned short cnt[RS_THREADS][RADIX];  // 32 KB of the 320 KB WGP LDS
  const int t = threadIdx.x, b = blockIdx.x;
  for (int d = 0; d < RADIX; ++d) cnt[t][d] = 0;
  const size_t base = (size_t)b * RS_CHUNK + (size_t)t * RS_ITEMS;
  for (int s = 0; s < RS_ITEMS; ++s) {
    uint32_t d = (uint32_t)((keys[base + s] >> shift) & 0xFF);
    cnt[t][d]++;
  }
  __syncthreads();
  // block total per digit; digit-major global layout => linear scan gives
  // (all smaller digits anywhere) + (same digit, earlier blocks)
  for (int k = 0; k < 4; ++k) {
    const int d = t * 4 + k;
    uint32_t sum = 0;
    for (int tt = 0; tt < RS_THREADS; ++tt) sum += cnt[tt][d];
    ghist[(size_t)d * RS_BLOCKS + b] = sum;
  }
}

__global__ __launch_bounds__(256) void k_scan(uint32_t* __restrict__ g)
{
  __shared__ uint32_t tmp[256];
  __shared__ uint32_t carry;
  const int t = threadIdx.x;
  if (t == 0) carry = 0;
  __syncthreads();
  const int total = RADIX * RS_BLOCKS;     // 64000, divisible by 256
  for (int off = 0; off < total; off += 256) {
    uint32_t v = g[off + t];
    tmp[t] = v;
    __syncthreads();
    for (int s = 1; s < 256; s <<= 1) {
      uint32_t a = (t >= s) ? tmp[t - s] : 0u;
      __syncthreads();
      tmp[t] += a;
      __syncthreads();
    }
    const uint32_t incl = tmp[t];
    const uint32_t c = carry;
    __syncthreads();
    g[off + t] = c + incl - v;             // exclusive
    if (t == 255) carry = c + incl;
    __syncthreads();
  }
}

__global__ __launch_bounds__(RS_THREADS) void k_scatter(
    const unsigned long long* __restrict__ kin, const uint32_t* __restrict__ vin,
    unsigned long long* __restrict__ kout, uint32_t* __restrict__ vout,
    const uint32_t* __restrict__ gbase, int shift)
{
  __shared__ unsigned short cnt[RS_THREADS][RADIX];  // 32 KB
  const int t = threadIdx.x, b = blockIdx.x;
  for (int d = 0; d < RADIX; ++d) cnt[t][d] = 0;
  const size_t base = (size_t)b * RS_CHUNK + (size_t)t * RS_ITEMS;
  for (int s = 0; s < RS_ITEMS; ++s) {
    uint32_t d = (uint32_t)((kin[base + s] >> shift) & 0xFF);
    cnt[t][d]++;
  }
  __syncthreads();
  // exclusive prefix across threads, per digit -> stable intra-block offsets
  for (int k = 0; k < 4; ++k) {
    const int d = t * 4 + k;
    uint32_t run = 0;
    for (int tt = 0; tt < RS_THREADS; ++tt) {
      uint32_t x = cnt[tt][d];
      cnt[tt][d] = (unsigned short)run;
      run += x;
    }
  }
  __syncthreads();
  // serial per-thread scatter: only thread t touches row t => no atomics, stable
  for (int s = 0; s < RS_ITEMS; ++s) {
    const unsigned long long kk = kin[base + s];
    const uint32_t d = (uint32_t)((kk >> shift) & 0xFF);
    const uint32_t pos = gbase[(size_t)d * RS_BLOCKS + b] + cnt[t][d];
    cnt[t][d]++;
    kout[pos] = kk;
    vout[pos] = vin[base + s];
  }
}

// =====================================================================
// Final: gather triangles by payload, compute uniq mask
// =====================================================================
__global__ __launch_bounds__(256) void k_final(
    const unsigned long long* __restrict__ keys, const uint32_t* __restrict__ vals,
    const int* __restrict__ tris, float* __restrict__ out)
{
  const int p = blockIdx.x * 256 + threadIdx.x;
  if (p >= MTRI) return;
  const unsigned long long k = keys[p];
  const uint32_t idx = vals[p];
  out[3 * (size_t)p + 0] = (float)tris[(size_t)idx * 3 + 0];
  out[3 * (size_t)p + 1] = (float)tris[(size_t)idx * 3 + 1];
  out[3 * (size_t)p + 2] = (float)tris[(size_t)idx * 3 + 2];
  bool u;
  if (k == SENTK)      u = false;
  else if (p == 0)     u = true;
  else                 u = (keys[p - 1] != k);
  out[(size_t)MTRI * 3 + p] = u ? 1.0f : 0.0f;
}

// =====================================================================
// Host launcher (graph-capture safe: only kernel launches on stream)
// =====================================================================
extern "C" void kernel_launch(void* const* d_in, const int* in_sizes, int n_in,
                              void* d_out, int out_size, void* d_ws, size_t ws_size,
                              hipStream_t stream)
{
  const float* points = (const float*)d_in[0];
  const float* logits = (const float*)d_in[1];
  const int*   knn    = (const int*)d_in[2];
  float* out = (float*)d_out;
  (void)in_sizes; (void)n_in; (void)out_size; (void)ws_size;

  // workspace carve-out (~58 MB total)
  char* ws = (char*)d_ws;
  size_t off = 0;
  auto carve = [&](size_t bytes) -> void* {
    void* p = (void*)(ws + off);
    off += (bytes + 255) & ~(size_t)255;
    return p;
  };
  unsigned long long* keysA = (unsigned long long*)carve((size_t)MTRI * 8);
  unsigned long long* keysB = (unsigned long long*)carve((size_t)MTRI * 8);
  uint32_t* valsA = (uint32_t*)carve((size_t)MTRI * 4);
  uint32_t* valsB = (uint32_t*)carve((size_t)MTRI * 4);
  int*      tris  = (int*)carve((size_t)MTRI * 3 * 4);
  uint32_t* ghist = (uint32_t*)carve((size_t)RADIX * RS_BLOCKS * 4);

  k_stage1<<<ROWS / 256, 256, 0, stream>>>(points, logits, knn, keysA, valsA, tris);

  unsigned long long *ka = keysA, *kb = keysB;
  uint32_t *va = valsA, *vb = valsB;
  for (int pass = 0; pass < PASSES; ++pass) {
    const int shift = pass * 8;
    k_hist   <<<RS_BLOCKS, RS_THREADS, 0, stream>>>(ka, ghist, shift);
    k_scan   <<<1, 256, 0, stream>>>(ghist);
    k_scatter<<<RS_BLOCKS, RS_THREADS, 0, stream>>>(ka, va, kb, vb, ghist, shift);
    unsigned long long* tk = ka; ka = kb; kb = tk;
    uint32_t* tv = va; va = vb; vb = tv;
  }
  // PASSES is even => sorted data ends in keysA/valsA (== ka/va either way)
  k_final<<<(MTRI + 255) / 256, 256, 0, stream>>>(ka, va, tris, out);
}